// SpikeLayer_83150566851380
// MI455X (gfx1250) — compile-verified
//
#include <hip/hip_runtime.h>
#include <stdint.h>

// SpikeLayer: inverse-CDF channel sampling.
//   x  : [B=32, C=512, H=32, W=32] f32 (non-negative)
//   rv : [B=32, S=512, H=32, W=32] f32 in [0,1)
//   out: [B, S, H, W] = first c with cdf[c] >= u  (written as f32 indices)
//
// One workgroup per (b,h) row of 32 pixels. The 512x32 f32 tile of x
// (rows of 128B at 4KB stride) is DMA'd into LDS by the Tensor Data Mover
// (tensor_load_to_lds, TENSORcnt), then scanned and binary-searched in LDS.

#define Bb 32
#define Cc 512
#define Hh 32
#define Ww 32
#define Ss 512

typedef __attribute__((ext_vector_type(4))) unsigned int u32x4;
typedef __attribute__((ext_vector_type(8))) int          i32x8;
typedef __attribute__((ext_vector_type(4))) int          i32x4;

__global__ __launch_bounds__(256) void spike_sample_kernel(
    const float* __restrict__ x,
    const float* __restrict__ rv,
    float* __restrict__ out) {
  // cdf tile: [c*32 + w], 64KB. ssum: per-pixel segment sums, stride 9 (bank-pad).
  __shared__ float lds[Cc * Ww + Ww * 9];
  float* cdf  = lds;
  float* ssum = lds + Cc * Ww;

  const int tid = threadIdx.x;
  const int w   = tid & 31;   // pixel within row
  const int r   = tid >> 5;   // segment / wave id (0..7)
  const int bh  = blockIdx.x;
  const int b   = bh >> 5;    // H == 32
  const int h   = bh & 31;

  // ---- Phase 0: TDM bulk copy of the 512x32 f32 tile, global -> LDS ----
  // Tile start in global memory: x + (b*C*H*W + h*W); row stride = H*W elems.
  if (r == 0) {  // one wave issues the descriptor (TDM ignores EXEC; wave op)
    uint64_t gaddr =
        (uint64_t)(const void*)(x + ((size_t)b * (Cc * Hh * Ww) + (size_t)h * Ww));

    u32x4 g0;
    g0[0] = 1u;                                   // count=1 (valid descriptor)
    g0[1] = 0u;                                   // lds_addr = 0 (start of our alloc)
    g0[2] = (unsigned int)gaddr;                  // global_addr[31:0]
    g0[3] = (unsigned int)((gaddr >> 32) & 0x01FFFFFFu) | 0x80000000u; // addr[56:32] | type=2

    i32x8 g1;
    g1[0] = (int)(2u << 16);      // workgroup_mask=0 | data_size=2 (4 bytes)
    g1[1] = (int)(32u << 16);     // atomic_barrier_addr=0 | tensor_dim0=32 (lo16)
    g1[2] = (int)(512u << 16);    // tensor_dim0 hi=0 | tensor_dim1=512 (lo16)
    g1[3] = (int)(32u << 16);     // tensor_dim1 hi=0 | tile_dim0=32
    g1[4] = 512;                  // tile_dim1=512 | tile_dim2=0
    g1[5] = 1024;                 // tensor_dim0_stride = H*W (lo32)
    g1[6] = 0;                    // stride hi=0 | tensor_dim1_stride lo=0
    g1[7] = 0;

    i32x4 gz4 = {0, 0, 0, 0};               // groups 2/3 unused for 2D tile
    i32x8 gz8 = {0, 0, 0, 0, 0, 0, 0, 0};   // trailing group (6-arg toolchain form)
    __builtin_amdgcn_tensor_load_to_lds(g0, g1, gz4, gz4, gz8, 0);
    __builtin_amdgcn_s_wait_tensorcnt(0);
  }
  __syncthreads();

  // ---- Phase 1: inclusive scan within each 64-channel segment (in place) ----
  const int base = (r << 6) * Ww + w;   // (r*64)*32 + w
  float acc = 0.0f;
  #pragma unroll 4
  for (int j = 0; j < 64; ++j) {
    const int idx = base + j * Ww;
    acc += cdf[idx];
    cdf[idx] = acc;
  }
  ssum[w * 9 + r] = acc;                // segment total
  __syncthreads();

  // ---- Phase 2: add exclusive cross-segment offset ----
  float off = 0.0f;
  #pragma unroll
  for (int k = 0; k < 8; ++k)
    off += (k < r) ? ssum[w * 9 + k] : 0.0f;
  if (r != 0) {
    #pragma unroll 4
    for (int j = 0; j < 64; ++j) cdf[base + j * Ww] += off;
  }
  __syncthreads();

  // ---- Phase 3: 64 inverse-CDF queries per thread ----
  // Top 3 binary-search levels cached in registers (probe ids fixed per level).
  const float total = cdf[511 * Ww + w];     // normalization folded into u
  const float c255  = cdf[255 * Ww + w];
  const float c127  = cdf[127 * Ww + w];
  const float c383  = cdf[383 * Ww + w];
  const float c63   = cdf[ 63 * Ww + w];
  const float c191  = cdf[191 * Ww + w];
  const float c319  = cdf[319 * Ww + w];
  const float c447  = cdf[447 * Ww + w];

  const size_t pixbase = (size_t)b * (Ss * Hh * Ww) + (size_t)h * Ww + (size_t)w;
  #pragma unroll 2
  for (int k = 0; k < 64; ++k) {
    const int s = (r << 6) + k;
    const size_t g = pixbase + (size_t)s * (Hh * Ww);  // lane-consecutive in w
    const float u = rv[g] * total;

    // lower_bound over 512 entries: pos = #elements < u
    int pos = (c255 < u) ? 256 : 0;
    const float l1 = pos ? c383 : c127;
    pos += (l1 < u) ? 128 : 0;
    const float lA = (pos & 256) ? c319 : c63;
    const float lB = (pos & 256) ? c447 : c191;
    const float l2 = (pos & 128) ? lB : lA;
    pos += (l2 < u) ? 64 : 0;
    #pragma unroll
    for (int step = 32; step > 0; step >>= 1) {
      const float v = cdf[(pos + step - 1) * Ww + w];  // conflict-free: bank=(32c+w)%64
      pos += (v < u) ? step : 0;
    }
    out[g] = (float)pos;
  }
}

extern "C" void kernel_launch(void* const* d_in, const int* in_sizes, int n_in,
                              void* d_out, int out_size, void* d_ws, size_t ws_size,
                              hipStream_t stream) {
  (void)in_sizes; (void)n_in; (void)out_size; (void)d_ws; (void)ws_size;
  const float* x  = (const float*)d_in[0];
  const float* rv = (const float*)d_in[1];
  float* out      = (float*)d_out;
  spike_sample_kernel<<<dim3(Bb * Hh), dim3(256), 0, stream>>>(x, rv, out);
}